// AttentionPooling_59450937311582
// MI455X (gfx1250) — compile-verified
//
#include <hip/hip_runtime.h>
#include <math.h>

typedef float v2f __attribute__((ext_vector_type(2)));
typedef float v8f __attribute__((ext_vector_type(8)));

#define DIM 256          // hidden dim (matches reference)
#define TPB 256          // threads per block (8 waves on wave32)
#define NEG_INF (-__builtin_inff())

__device__ __forceinline__ float blockReduceMax(float v, float* sdata) {
    int tid = threadIdx.x;
    sdata[tid] = v;
    __syncthreads();
    for (int off = TPB / 2; off > 0; off >>= 1) {
        if (tid < off) sdata[tid] = fmaxf(sdata[tid], sdata[tid + off]);
        __syncthreads();
    }
    float r = sdata[0];
    __syncthreads();
    return r;
}

__device__ __forceinline__ float blockReduceSum(float v, float* sdata) {
    int tid = threadIdx.x;
    sdata[tid] = v;
    __syncthreads();
    for (int off = TPB / 2; off > 0; off >>= 1) {
        if (tid < off) sdata[tid] += sdata[tid + off];
        __syncthreads();
    }
    float r = sdata[0];
    __syncthreads();
    return r;
}

// ---------------------------------------------------------------------------
// K1: s[i] = dot(x[i,:], W) + b  via V_WMMA_F32_16X16X4_F32.
// Each wave handles 16 rows; B operand = W chunk broadcast to all 16 columns.
// Also emits per-block max of s for the global softmax.
// ---------------------------------------------------------------------------
__global__ void score_wmma_kernel(const float* __restrict__ x,
                                  const float* __restrict__ W,
                                  const float* __restrict__ bias,
                                  float* __restrict__ s,
                                  float* __restrict__ bmax,
                                  int N) {
    __shared__ float sdata[TPB];
    const int lane = threadIdx.x & 31;
    const int wave = threadIdx.x >> 5;
    const int half = lane >> 4;      // lane half selects K pair (ISA 16x4 f32 layout)
    const int m    = lane & 15;      // row within tile
    const int rowBase = (blockIdx.x * 8 + wave) * 16;

    int loadRow = rowBase + m;
    if (loadRow >= N) loadRow = N - 1;              // clamp (keeps EXEC uniform)
    const float* xrow = x + (size_t)loadRow * DIM + 2 * half;
    const float* wptr = W + 2 * half;

    v8f c = {0.f, 0.f, 0.f, 0.f, 0.f, 0.f, 0.f, 0.f};
    #pragma unroll 4
    for (int k0 = 0; k0 < DIM; k0 += 4) {
        v2f a  = *(const v2f*)(xrow + k0);          // A[m, 2h+v], v=0..1
        v2f bv = *(const v2f*)(wptr + k0);          // B[2h+v, all n] = W[k0+2h+v]
        c = __builtin_amdgcn_wmma_f32_16x16x4_f32(
                /*neg_a=*/false, a, /*neg_b=*/false, bv,
                /*c_mod=*/(short)0, c, /*reuse_a=*/false, /*reuse_b=*/false);
    }

    const float bval = bias[0];
    float lmax = NEG_INF;
    if (m == 0) {   // lane 0 holds rows 0..7, lane 16 holds rows 8..15 (col 0)
        #pragma unroll
        for (int r = 0; r < 8; ++r) {
            int row = rowBase + half * 8 + r;
            if (row < N) {
                float val = c[r] + bval;
                s[row] = val;
                lmax = fmaxf(lmax, val);
            }
        }
    }
    float bm = blockReduceMax(lmax, sdata);
    if (threadIdx.x == 0) bmax[blockIdx.x] = bm;
}

// K2: single-block max reduction -> gmax
__global__ void reduce_max_kernel(const float* __restrict__ in, int n,
                                  float* __restrict__ out) {
    __shared__ float sdata[TPB];
    float v = NEG_INF;
    for (int i = threadIdx.x; i < n; i += TPB) v = fmaxf(v, in[i]);
    float r = blockReduceMax(v, sdata);
    if (threadIdx.x == 0) out[0] = r;
}

// K3: p[i] = exp(s[i] - gmax) in place; per-block partial sums
__global__ void exp_kernel(float* __restrict__ s, const float* __restrict__ gmax,
                           float* __restrict__ bsum, int N) {
    __shared__ float sdata[TPB];
    int i = blockIdx.x * TPB + threadIdx.x;
    float v = 0.f;
    if (i < N) {
        v = expf(s[i] - gmax[0]);
        s[i] = v;
    }
    float r = blockReduceSum(v, sdata);
    if (threadIdx.x == 0) bsum[blockIdx.x] = r;
}

// K4: single-block sum reduction -> gsum
__global__ void reduce_sum_kernel(const float* __restrict__ in, int n,
                                  float* __restrict__ out) {
    __shared__ float sdata[TPB];
    float v = 0.f;
    for (int i = threadIdx.x; i < n; i += TPB) v += in[i];
    float r = blockReduceSum(v, sdata);
    if (threadIdx.x == 0) out[0] = r;
}

// ---------------------------------------------------------------------------
// K5: one block per segment (batch_idx is sorted -> binary search row range).
// g = p / gsum; segment softmax of g; pooled[seg,:] = sum_i w_i * x[i,:].
// 256 threads <-> 256 columns, no atomics, coalesced x reads.
// ---------------------------------------------------------------------------
__global__ void segment_pool_kernel(const float* __restrict__ x,
                                    const int* __restrict__ idx,
                                    const float* __restrict__ p,
                                    const float* __restrict__ gsum_ptr,
                                    float* __restrict__ out,
                                    int N) {
    __shared__ float sdata[TPB];
    __shared__ float wbuf[TPB];
    __shared__ int sbounds[2];
    const int seg = blockIdx.x;
    const int tid = threadIdx.x;

    if (tid == 0) {
        int lo = 0, hi = N;
        while (lo < hi) { int mid = (lo + hi) >> 1; if (idx[mid] < seg) lo = mid + 1; else hi = mid; }
        int start = lo;
        hi = N;
        while (lo < hi) { int mid = (lo + hi) >> 1; if (idx[mid] < seg + 1) lo = mid + 1; else hi = mid; }
        sbounds[0] = start;
        sbounds[1] = lo;
    }
    __syncthreads();
    const int start = sbounds[0];
    const int end   = sbounds[1];

    if (start >= end) {                       // empty segment -> zeros
        out[(size_t)seg * DIM + tid] = 0.f;
        return;
    }

    const float inv_gsum = 1.0f / gsum_ptr[0];

    // segment max of g (= max(p) * inv_gsum since inv_gsum > 0)
    float lmax = NEG_INF;
    for (int r = start + tid; r < end; r += TPB) lmax = fmaxf(lmax, p[r]);
    const float segmax = blockReduceMax(lmax, sdata) * inv_gsum;

    // denom = sum exp(g - segmax)
    float lsum = 0.f;
    for (int r = start + tid; r < end; r += TPB)
        lsum += expf(p[r] * inv_gsum - segmax);
    const float inv_denom = 1.0f / blockReduceSum(lsum, sdata);

    // weighted column sum; w-values staged through LDS per 256-row chunk
    float acc = 0.f;
    for (int base = start; base < end; base += TPB) {
        int cnt = min(TPB, end - base);
        float wv = 0.f;
        if (tid < cnt)
            wv = expf(p[base + tid] * inv_gsum - segmax) * inv_denom;
        wbuf[tid] = wv;
        __syncthreads();
        for (int j = 0; j < cnt; ++j)
            acc += wbuf[j] * x[(size_t)(base + j) * DIM + tid];
        __syncthreads();
    }
    out[(size_t)seg * DIM + tid] = acc;
}

// ---------------------------------------------------------------------------
extern "C" void kernel_launch(void* const* d_in, const int* in_sizes, int n_in,
                              void* d_out, int out_size, void* d_ws, size_t ws_size,
                              hipStream_t stream) {
    const float* x    = (const float*)d_in[0];
    const int*   idx  = (const int*)d_in[1];
    const float* W    = (const float*)d_in[2];
    const float* bias = (const float*)d_in[3];

    const int N    = in_sizes[1];          // batch_idx has N entries
    const int nseg = out_size / DIM;       // B segments

    const int nb1 = (N + 127) / 128;       // score kernel: 128 rows / block
    const int nb3 = (N + TPB - 1) / TPB;   // exp kernel blocks

    float* ws   = (float*)d_ws;
    float* s    = ws;                      // N floats: scores -> exp(s-gmax)
    float* bmax = s + N;                   // nb1 floats
    float* bsum = bmax + nb1;              // nb3 floats
    float* gmax = bsum + nb3;              // 1 float
    float* gsum = gmax + 1;                // 1 float

    score_wmma_kernel<<<nb1, TPB, 0, stream>>>(x, W, bias, s, bmax, N);
    reduce_max_kernel<<<1, TPB, 0, stream>>>(bmax, nb1, gmax);
    exp_kernel<<<nb3, TPB, 0, stream>>>(s, gmax, bsum, N);
    reduce_sum_kernel<<<1, TPB, 0, stream>>>(bsum, nb3, gsum);
    segment_pool_kernel<<<nseg, TPB, 0, stream>>>(x, idx, s, gsum, (float*)d_out, N);
}